// TrackEmbeddingHead_40931038331138
// MI455X (gfx1250) — compile-verified
//
#include <hip/hip_runtime.h>
#include <math.h>

// ---- problem constants (match reference) ----
#define Bn     8
#define Nn     64
#define Cc     256      // input channels
#define Hh     100
#define Ww     100
#define HWp    10000    // H*W
#define PROJ   256
#define HIDD   128
#define EMB    64
#define NTILES 625      // HW/16
#define MQUADS 4        // PROJ/64 : each wave owns 64 output rows (4 wmma tiles)

typedef __attribute__((ext_vector_type(2))) float v2f;
typedef __attribute__((ext_vector_type(8))) float v8f;

// ---------------------------------------------------------------------------
// Kernel 1: 1x1 conv as per-batch GEMM with V_WMMA_F32_16X16X4_F32.
// Each wave32 computes a 64(M=out ch) x 16(N=pixel) tile: 4 accumulators
// sharing one B fragment per K-step (B loads are the strided/expensive ones).
// Also emits per-tile (sum, sumsq) partials for GroupNorm (deterministic).
// ---------------------------------------------------------------------------
__global__ __launch_bounds__(256)
void conv_wmma_kernel(const float* __restrict__ feat, const float* __restrict__ cw,
                      float* __restrict__ y, float2* __restrict__ part) {
  int wave = (blockIdx.x * blockDim.x + threadIdx.x) >> 5;
  int lane = threadIdx.x & 31;
  int nt = wave % NTILES;
  int mq = (wave / NTILES) % MQUADS;
  int b  = wave / (NTILES * MQUADS);
  int lm = lane & 15;   // M (for A) / N (for B,C,D) within tile
  int hi = lane >> 4;   // lane-half select

  const float* fb   = feat + (size_t)b * Cc * HWp;
  int Mbase = mq * 64, Nbase = nt * 16;
  // A: conv_w[o, c], row-major [PROJ, C].  lane half selects K pair {0,1} vs {2,3}.
  const float* arow = cw + (Mbase + lm) * Cc + hi * 2;
  // B: features[b, c, hw].  lane half selects K rows {0,1} vs {2,3}; lm = column.
  const float* bcol = fb + (size_t)(hi * 2) * HWp + (Nbase + lm);

  v8f acc0 = {0.f,0.f,0.f,0.f,0.f,0.f,0.f,0.f};
  v8f acc1 = acc0, acc2 = acc0, acc3 = acc0;

  for (int k0 = 0; k0 < Cc; k0 += 4) {
    v2f bv;
    bv.x = bcol[(size_t)k0 * HWp];
    bv.y = bcol[(size_t)(k0 + 1) * HWp];
    v2f a0, a1, a2, a3;
    a0.x = arow[k0];              a0.y = arow[k0 + 1];
    a1.x = arow[16 * Cc + k0];    a1.y = arow[16 * Cc + k0 + 1];
    a2.x = arow[32 * Cc + k0];    a2.y = arow[32 * Cc + k0 + 1];
    a3.x = arow[48 * Cc + k0];    a3.y = arow[48 * Cc + k0 + 1];
    // 8 args: (neg_a, A, neg_b, B, c_mod, C, reuse_a, reuse_b)
    acc0 = __builtin_amdgcn_wmma_f32_16x16x4_f32(false, a0, false, bv, (short)0, acc0, false, false);
    acc1 = __builtin_amdgcn_wmma_f32_16x16x4_f32(false, a1, false, bv, (short)0, acc1, false, false);
    acc2 = __builtin_amdgcn_wmma_f32_16x16x4_f32(false, a2, false, bv, (short)0, acc2, false, false);
    acc3 = __builtin_amdgcn_wmma_f32_16x16x4_f32(false, a3, false, bv, (short)0, acc3, false, false);
  }

  // Store D per ISA C/D layout; accumulate tile sum / sumsq for GN stats.
  float* yb = y + (size_t)b * PROJ * HWp;
  int col = Nbase + lm;
  float s = 0.f, sq = 0.f;
  v8f accs[4] = {acc0, acc1, acc2, acc3};
  #pragma unroll
  for (int i = 0; i < 4; ++i) {
    #pragma unroll
    for (int v = 0; v < 8; ++v) {
      int row = Mbase + i * 16 + v + hi * 8;
      float d = accs[i][v];
      yb[(size_t)row * HWp + col] = d;
      s += d; sq += d * d;
    }
  }
  // wave32 cross-lane reduce
  for (int off = 16; off > 0; off >>= 1) {
    s  += __shfl_xor(s,  off, 32);
    sq += __shfl_xor(sq, off, 32);
  }
  if (lane == 0)
    part[(size_t)b * (NTILES * MQUADS) + mq * NTILES + nt] = make_float2(s, sq);
}

// ---------------------------------------------------------------------------
// Kernel 2: reduce per-wave partials -> per-batch mean & rsqrt(var + eps)
// ---------------------------------------------------------------------------
__global__ __launch_bounds__(256)
void stats_kernel(const float2* __restrict__ part, float2* __restrict__ stats) {
  __shared__ float sh_s[256], sh_q[256];
  int b = blockIdx.x, t = threadIdx.x;
  const float2* p = part + (size_t)b * (NTILES * MQUADS);
  float s = 0.f, q = 0.f;
  for (int i = t; i < NTILES * MQUADS; i += 256) { float2 v = p[i]; s += v.x; q += v.y; }
  sh_s[t] = s; sh_q[t] = q; __syncthreads();
  for (int st = 128; st > 0; st >>= 1) {
    if (t < st) { sh_s[t] += sh_s[t + st]; sh_q[t] += sh_q[t + st]; }
    __syncthreads();
  }
  if (t == 0) {
    float cnt  = (float)PROJ * (float)HWp;
    float mean = sh_s[0] / cnt;
    float var  = sh_q[0] / cnt - mean * mean;
    stats[b] = make_float2(mean, rsqrtf(var + 1e-5f));
  }
}

// ---------------------------------------------------------------------------
// Kernel 3: per (b,c) slice: GroupNorm affine + SiLU, then 2D inclusive
// prefix sum (summed-area table) in LDS; written back in place.
// ---------------------------------------------------------------------------
__global__ __launch_bounds__(128)
void gn_silu_sat_kernel(float* __restrict__ y, const float2* __restrict__ stats,
                        const float* __restrict__ gw, const float* __restrict__ gb) {
  __shared__ float tile[HWp];          // 40 KB of the 320 KB/WGP LDS
  int b = blockIdx.x >> 8;
  int c = blockIdx.x & 255;
  int t = threadIdx.x;
  float2 st = stats[b];
  float gamma = gw[c], beta = gb[c];
  float* slice = y + ((size_t)b * PROJ + c) * HWp;

  for (int i = t; i < HWp; i += 128) {
    float v = (slice[i] - st.x) * st.y * gamma + beta;
    tile[i] = v / (1.0f + expf(-v));   // SiLU
  }
  __syncthreads();
  if (t < Hh) {                        // row prefix
    float run = 0.f; int base = t * Ww;
    for (int w = 0; w < Ww; ++w) { run += tile[base + w]; tile[base + w] = run; }
  }
  __syncthreads();
  if (t < Ww) {                        // column prefix
    float run = 0.f;
    for (int h = 0; h < Hh; ++h) { run += tile[h * Ww + t]; tile[h * Ww + t] = run; }
  }
  __syncthreads();
  for (int i = t; i < HWp; i += 128) slice[i] = tile[i];
}

// ---------------------------------------------------------------------------
// Kernel 4: one block per (b,n) box: 4-tap SAT pooling per channel,
// then MLP (256->128 SiLU ->64) + L2 normalize. Writes pooled/raw/emb.
// ---------------------------------------------------------------------------
__global__ __launch_bounds__(256)
void pool_mlp_kernel(const float* __restrict__ sat, const float* __restrict__ boxes,
                     const float* __restrict__ w1, const float* __restrict__ b1,
                     const float* __restrict__ w2, const float* __restrict__ b2,
                     float* __restrict__ out) {
  __shared__ float sp[PROJ];
  __shared__ float shid[HIDD];
  __shared__ float sraw[EMB];
  int bn = blockIdx.x;                 // b*64 + n
  int b  = bn >> 6;
  int t  = threadIdx.x;

  const float* bx = boxes + (size_t)bn * 4;
  float c0 = bx[0], c1 = bx[1], c2 = bx[2], c3 = bx[3];
  float x1f = fminf(c0, c2) * (float)Ww, x2f = fmaxf(c0, c2) * (float)Ww;
  float y1f = fminf(c1, c3) * (float)Hh, y2f = fmaxf(c1, c3) * (float)Hh;
  x1f = fminf(fmaxf(floorf(fminf(fmaxf(x1f, 0.f), (float)Ww)), 0.f), (float)Ww);
  x2f = fminf(fmaxf(ceilf (fminf(fmaxf(x2f, 0.f), (float)Ww)), 0.f), (float)Ww);
  y1f = fminf(fmaxf(floorf(fminf(fmaxf(y1f, 0.f), (float)Hh)), 0.f), (float)Hh);
  y2f = fminf(fmaxf(ceilf (fminf(fmaxf(y2f, 0.f), (float)Hh)), 0.f), (float)Hh);
  int ix1 = (int)x1f, ix2 = (int)x2f, iy1 = (int)y1f, iy2 = (int)y2f;
  int cnt = (iy2 - iy1) * (ix2 - ix1);

  {   // pooled value for channel t via 4 SAT taps
    const float* S = sat + ((size_t)b * PROJ + t) * HWp;
    float pv = 0.f;
    if (cnt > 0) {
      int r2 = iy2 - 1, q2 = ix2 - 1;
      float sum = S[r2 * Ww + q2];
      if (iy1 > 0)             sum -= S[(iy1 - 1) * Ww + q2];
      if (ix1 > 0)             sum -= S[r2 * Ww + (ix1 - 1)];
      if (iy1 > 0 && ix1 > 0)  sum += S[(iy1 - 1) * Ww + (ix1 - 1)];
      pv = sum / (float)cnt;
    }
    sp[t] = pv;
    out[65536 + (size_t)bn * PROJ + t] = pv;        // pooled output
  }
  __syncthreads();

  if (t < HIDD) {
    float acc = b1[t];
    for (int c = 0; c < PROJ; ++c) acc += sp[c] * w1[c * HIDD + t];
    shid[t] = acc / (1.0f + expf(-acc));            // SiLU
  }
  __syncthreads();

  if (t < EMB) {
    float acc = b2[t];
    for (int j = 0; j < HIDD; ++j) acc += shid[j] * w2[j * EMB + t];
    sraw[t] = acc;
    out[32768 + (size_t)bn * EMB + t] = acc;        // raw output
  }
  __syncthreads();

  if (t < EMB) {
    float s = 0.f;
    for (int k = 0; k < EMB; ++k) s += sraw[k] * sraw[k];
    float nrm = fmaxf(sqrtf(s), 1e-6f);
    out[(size_t)bn * EMB + t] = sraw[t] / nrm;      // emb output
  }
}

// ---------------------------------------------------------------------------
extern "C" void kernel_launch(void* const* d_in, const int* in_sizes, int n_in,
                              void* d_out, int out_size, void* d_ws, size_t ws_size,
                              hipStream_t stream) {
  const float* feat  = (const float*)d_in[0];
  const float* boxes = (const float*)d_in[1];
  const float* cw    = (const float*)d_in[2];
  const float* gw    = (const float*)d_in[3];
  const float* gb    = (const float*)d_in[4];
  const float* w1    = (const float*)d_in[5];
  const float* b1    = (const float*)d_in[6];
  const float* w2    = (const float*)d_in[7];
  const float* b2    = (const float*)d_in[8];
  float* out = (float*)d_out;

  char*   ws     = (char*)d_ws;
  float*  y      = (float*)ws;                                  // 81.92 MB
  size_t  ybytes = (size_t)Bn * PROJ * HWp * sizeof(float);
  float2* part   = (float2*)(ws + ybytes);                      // 160 KB
  size_t  pbytes = (size_t)Bn * NTILES * MQUADS * sizeof(float2);
  float2* stats  = (float2*)(ws + ybytes + pbytes);             // 64 B

  int waves = Bn * MQUADS * NTILES;                             // 20000
  conv_wmma_kernel<<<waves / 8, 256, 0, stream>>>(feat, cw, y, part);
  stats_kernel    <<<Bn,        256, 0, stream>>>(part, stats);
  gn_silu_sat_kernel<<<Bn * PROJ, 128, 0, stream>>>(y, stats, gw, gb);
  pool_mlp_kernel <<<Bn * Nn,   256, 0, stream>>>(y, boxes, w1, b1, w2, b2, out);
}